// GeometricProduct_8186207666926
// MI455X (gfx1250) — compile-verified
//
#include <hip/hip_runtime.h>
#include <hip/hip_bf16.h>

typedef __attribute__((ext_vector_type(2))) float v2f;
typedef __attribute__((ext_vector_type(8))) float v8f;

// ---------------------------------------------------------------------------
// Compile-time Cayley table for Cl(3,0), blade order 1,e1,e2,e3,e12,e13,e23,e123
// p = i*8 + j enumerates (i,j) pairs; k[p] = output blade, s[p] = +/-1.
// ---------------------------------------------------------------------------
struct CayleyTab {
    int   k[64];
    float s[64];
    int   bi[64];
    int   bj[64];
    constexpr CayleyTab() : k{}, s{}, bi{}, bj{} {
        const int order[8] = {0, 1, 2, 4, 3, 5, 6, 7};   // bitmask per blade slot
        int idx[8] = {0, 0, 0, 0, 0, 0, 0, 0};
        for (int t = 0; t < 8; ++t) idx[order[t]] = t;
        for (int a = 0; a < 8; ++a) {
            for (int b = 0; b < 8; ++b) {
                const int A = order[a], B = order[b];
                int sw = 0;
                int t = A >> 1;
                while (t) {                      // canonical reorder swap count
                    int x = t & B;
                    while (x) { sw += x & 1; x >>= 1; }
                    t >>= 1;
                }
                const int p = a * 8 + b;
                k[p]  = idx[A ^ B];
                s[p]  = (sw & 1) ? -1.0f : 1.0f;
                bi[p] = a;
                bj[p] = b;
            }
        }
    }
};
constexpr CayleyTab CAY{};

__device__ __forceinline__ float lane_bcast(float v, int srcLane) {
    // wave32 broadcast via ds_bpermute (byte index = lane*4)
    int r = __builtin_amdgcn_ds_bpermute(srcLane << 2, __builtin_bit_cast(int, v));
    return __builtin_bit_cast(float, r);
}

// ---------------------------------------------------------------------------
// y[bt,k,d] = sum_p  Ccoef[k,p] * x[bt,i(p),d] * x[bt,j(p),d]
// Formulated as D(16x16) = A(16x64) x B(64x16) chained over 16 WMMA f32 16x16x4
// steps; M rows = blades (0..7 used), N cols = 16 consecutive d-points.
// Each wave handles 2 tiles (32 points) per iteration for full coalescing.
// ---------------------------------------------------------------------------
__global__ __launch_bounds__(256) void
geoprod_wmma_kernel(const float* __restrict__ x, const float* __restrict__ w,
                    float* __restrict__ out, int BT) {
    const int lane = threadIdx.x & 31;
    const int m    = lane & 15;        // A-matrix row / column-point-in-tile
    const int hh   = lane >> 4;        // lane half (selects K pair within step)

    const int waveId = (int)((blockIdx.x * blockDim.x + threadIdx.x) >> 5);
    const int nWaves = (int)((gridDim.x * blockDim.x) >> 5);

    // ---- Build A = sign * sigmoid(w) in WMMA 16x4-per-step layout (once) ----
    // Step t covers pairs p = 4t..4t+3.
    //   a0 (VGPR0): lanes 0-15 -> C[m][4t+0], lanes 16-31 -> C[m][4t+2]
    //   a1 (VGPR1): lanes 0-15 -> C[m][4t+1], lanes 16-31 -> C[m][4t+3]
    float a0[16], a1[16];
#pragma unroll
    for (int t = 0; t < 16; ++t) {
        float sg[4];
#pragma unroll
        for (int q = 0; q < 4; ++q) {
            const int p = 4 * t + q;
            const float wv = w[CAY.bi[p] * 8 + CAY.bj[p]];   // uniform -> s_load
            sg[q] = CAY.s[p] / (1.0f + __builtin_expf(-wv)); // sign * sigmoid
        }
        const float lo0 = (m == CAY.k[4 * t + 0]) ? sg[0] : 0.0f;
        const float hi0 = (m == CAY.k[4 * t + 2]) ? sg[2] : 0.0f;
        const float lo1 = (m == CAY.k[4 * t + 1]) ? sg[1] : 0.0f;
        const float hi1 = (m == CAY.k[4 * t + 3]) ? sg[3] : 0.0f;
        a0[t] = hh ? hi0 : lo0;
        a1[t] = hh ? hi1 : lo1;
    }

    const int totalTiles = BT * 8;          // D=256 -> 8 tiles of 32 points / bt
    for (int tile = waveId; tile < totalTiles; tile += nWaves) {
        const int bt = tile >> 3;
        const int d0 = (tile & 7) << 5;
        const float* __restrict__ xp = x + (size_t)bt * 2048 + d0;

        // Coalesced load: lane owns point d0+lane, blades i=0..7 (128B/blade).
        float xl[8];
#pragma unroll
        for (int i = 0; i < 8; ++i) xl[i] = xp[(size_t)i * 256 + lane];

        // Distribute: xs0 = blades of tile-0 column point (m),
        //             xs1 = blades of tile-1 column point (16+m).
        float xs0[8], xs1[8];
#pragma unroll
        for (int i = 0; i < 8; ++i) {
            xs0[i] = lane_bcast(xl[i], m);
            xs1[i] = lane_bcast(xl[i], 16 + m);
        }

        v8f acc0 = {0.f, 0.f, 0.f, 0.f, 0.f, 0.f, 0.f, 0.f};
        v8f acc1 = {0.f, 0.f, 0.f, 0.f, 0.f, 0.f, 0.f, 0.f};

#pragma unroll
        for (int t = 0; t < 16; ++t) {
            v2f A;
            A.x = a0[t];
            A.y = a1[t];

            // B (4x16 f32): lanes 0-15 hold K=4t,4t+1; lanes 16-31 hold 4t+2,4t+3
            const int p0 = 4 * t;
            // tile 0 pair-products
            const float q0 = xs0[CAY.bi[p0 + 0]] * xs0[CAY.bj[p0 + 0]];
            const float q1 = xs0[CAY.bi[p0 + 1]] * xs0[CAY.bj[p0 + 1]];
            const float q2 = xs0[CAY.bi[p0 + 2]] * xs0[CAY.bj[p0 + 2]];
            const float q3 = xs0[CAY.bi[p0 + 3]] * xs0[CAY.bj[p0 + 3]];
            v2f B0;
            B0.x = hh ? q2 : q0;
            B0.y = hh ? q3 : q1;
            acc0 = __builtin_amdgcn_wmma_f32_16x16x4_f32(
                false, A, false, B0, (short)0, acc0, false, false);

            // tile 1 pair-products (independent chain -> pipelines)
            const float r0 = xs1[CAY.bi[p0 + 0]] * xs1[CAY.bj[p0 + 0]];
            const float r1 = xs1[CAY.bi[p0 + 1]] * xs1[CAY.bj[p0 + 1]];
            const float r2 = xs1[CAY.bi[p0 + 2]] * xs1[CAY.bj[p0 + 2]];
            const float r3 = xs1[CAY.bi[p0 + 3]] * xs1[CAY.bj[p0 + 3]];
            v2f B1;
            B1.x = hh ? r2 : r0;
            B1.y = hh ? r3 : r1;
            acc1 = __builtin_amdgcn_wmma_f32_16x16x4_f32(
                false, A, false, B1, (short)0, acc1, false, false);
        }

        // Store: D row r = blade k=r. Lanes 0-15 take tile0 (points d0+lane),
        // lanes 16-31 pull tile1 row from lane (lane-16) -> full 128B store.
        float* __restrict__ op = out + (size_t)bt * 2048 + d0 + lane;
#pragma unroll
        for (int r = 0; r < 8; ++r) {
            const float hiVal = lane_bcast(acc1[r], m);
            const float v = (lane < 16) ? acc0[r] : hiVal;
            op[(size_t)r * 256] = v;
        }
    }
}

extern "C" void kernel_launch(void* const* d_in, const int* in_sizes, int n_in,
                              void* d_out, int out_size, void* d_ws, size_t ws_size,
                              hipStream_t stream) {
    const float* x   = (const float*)d_in[0];             // (B,T,8,256) f32
    const float* w   = (const float*)d_in[1];             // (8,8) f32
    float*       out = (float*)d_out;                     // (B,T,8,256) f32

    const int BT = in_sizes[0] / (8 * 256);               // 8192
    const int totalTiles = BT * 8;                        // 32-point tiles

    int blocks = 1024;                                    // 8192 waves, 8 tiles/wave
    const int maxBlocks = (totalTiles + 7) / 8;           // 8 waves per block
    if (blocks > maxBlocks) blocks = maxBlocks;
    if (blocks < 1) blocks = 1;

    geoprod_wmma_kernel<<<blocks, 256, 0, stream>>>(x, w, out, BT);
}